// InverseWaveletTransform_77395310674220
// MI455X (gfx1250) — compile-verified
//
#include <hip/hip_runtime.h>
#include <cstdint>

// Inverse 2D Haar DWT, shapes fixed by the reference:
//   inputs : 4 x [8,64,128,128] f32  (A, H, V, D)
//   output :     [8,64,256,256] f32
//
// Memory-bound (~268 MB @ 23.3 TB/s => ~11.5 us floor). Strategy:
//   - all global traffic as b128 (each thread: 4 input float4 loads, 4 output float4 stores)
//   - inputs staged via CDNA5 async global->LDS b128 copies (ASYNCcnt path)
//   - output written with non-temporal hint (write-once stream, keep L2 for reads)

typedef float f4 __attribute__((ext_vector_type(4)));

#define BLOCK 256

__global__ __launch_bounds__(BLOCK) void haar_idwt2_b128_async(
    const float* __restrict__ A,
    const float* __restrict__ H,
    const float* __restrict__ V,
    const float* __restrict__ D,
    float* __restrict__ out,
    unsigned total_threads)
{
    // SoA staging: lane-contiguous 16B chunks -> conflict-free ds_read_b128
    __shared__ f4 stage[4][BLOCK];

    const unsigned tid = threadIdx.x;
    const unsigned gid = blockIdx.x * BLOCK + tid;
    if (gid >= total_threads) return;

    const unsigned colv = gid & 31u;   // which float4 within a 128-wide input row
    const unsigned row  = gid >> 5;    // flattened (b*C + c)*Hc + hc
    const unsigned hc   = row & 127u;  // Hc = 128
    const unsigned bc   = row >> 7;    // b*C + c  (0..511)

    const unsigned in_off  = row * 128u + colv * 4u;                 // elements
    const unsigned out_off = bc * 65536u + hc * 512u + colv * 8u;    // elements

    // Global source addresses (64-bit) and LDS destination byte offsets (32-bit)
    const unsigned long long gA = (unsigned long long)(uintptr_t)(A + in_off);
    const unsigned long long gH = (unsigned long long)(uintptr_t)(H + in_off);
    const unsigned long long gV = (unsigned long long)(uintptr_t)(V + in_off);
    const unsigned long long gD = (unsigned long long)(uintptr_t)(D + in_off);

    const unsigned lA = (unsigned)(uintptr_t)(&stage[0][tid]);
    const unsigned lH = (unsigned)(uintptr_t)(&stage[1][tid]);
    const unsigned lV = (unsigned)(uintptr_t)(&stage[2][tid]);
    const unsigned lD = (unsigned)(uintptr_t)(&stage[3][tid]);

    // CDNA5 async global -> LDS copies (tracked by ASYNCcnt)
    asm volatile("global_load_async_to_lds_b128 %0, %1, off" :: "v"(lA), "v"(gA) : "memory");
    asm volatile("global_load_async_to_lds_b128 %0, %1, off" :: "v"(lH), "v"(gH) : "memory");
    asm volatile("global_load_async_to_lds_b128 %0, %1, off" :: "v"(lV), "v"(gV) : "memory");
    asm volatile("global_load_async_to_lds_b128 %0, %1, off" :: "v"(lD), "v"(gD) : "memory");
    asm volatile("s_wait_asynccnt 0" ::: "memory");

    const f4 a = stage[0][tid];
    const f4 h = stage[1][tid];
    const f4 v = stage[2][tid];
    const f4 d = stage[3][tid];

    // Butterfly per column:
    //   t0=a+d, t1=a-d, t2=h+v, t3=h-v
    //   x00=(t0+t2)/2  x01=(t1+t3)/2  x10=(t1-t3)/2  x11=(t0-t2)/2
    f4 top0, top1, bot0, bot1;
#pragma unroll
    for (int j = 0; j < 4; ++j) {
        const float t0 = a[j] + d[j];
        const float t1 = a[j] - d[j];
        const float t2 = h[j] + v[j];
        const float t3 = h[j] - v[j];
        const float x00 = (t0 + t2) * 0.5f;
        const float x01 = (t1 + t3) * 0.5f;
        const float x10 = (t1 - t3) * 0.5f;
        const float x11 = (t0 - t2) * 0.5f;
        if (j < 2) {
            top0[2 * j]     = x00;  top0[2 * j + 1] = x01;
            bot0[2 * j]     = x10;  bot0[2 * j + 1] = x11;
        } else {
            top1[2 * (j - 2)]     = x00;  top1[2 * (j - 2) + 1] = x01;
            bot1[2 * (j - 2)]     = x10;  bot1[2 * (j - 2) + 1] = x11;
        }
    }

    // Write-once output: non-temporal b128 stores (row 2*hc and row 2*hc+1)
    __builtin_nontemporal_store(top0, (f4*)(out + out_off));
    __builtin_nontemporal_store(top1, (f4*)(out + out_off + 4));
    __builtin_nontemporal_store(bot0, (f4*)(out + out_off + 256));
    __builtin_nontemporal_store(bot1, (f4*)(out + out_off + 260));
}

extern "C" void kernel_launch(void* const* d_in, const int* in_sizes, int n_in,
                              void* d_out, int out_size, void* d_ws, size_t ws_size,
                              hipStream_t stream) {
    (void)n_in; (void)out_size; (void)d_ws; (void)ws_size;
    const float* A = (const float*)d_in[0];  // approximation
    const float* H = (const float*)d_in[1];  // detail_h
    const float* V = (const float*)d_in[2];  // detail_v
    const float* D = (const float*)d_in[3];  // detail_d
    float* out = (float*)d_out;

    const unsigned n       = (unsigned)in_sizes[0];  // 8*64*128*128 = 8388608
    const unsigned threads = n / 4u;                 // one float4 column-group per thread
    const unsigned blocks  = (threads + BLOCK - 1) / BLOCK;

    haar_idwt2_b128_async<<<blocks, BLOCK, 0, stream>>>(A, H, V, D, out, threads);
}